// MultiHeadAttentionLayer_79645873537180
// MI455X (gfx1250) — compile-verified
//
#include <hip/hip_runtime.h>
#include <hip/hip_bf16.h>

#define IN_DIM   128
#define OUT_DIM  128   // HEADS * HEAD_DIM
#define HEADS    8
#define HEAD_DIM 16
#define ROWS_PER_BLOCK 64   // 4 stacked 16-row WMMA tiles -> 4x weight reuse

typedef __attribute__((ext_vector_type(2))) float v2f;
typedef __attribute__((ext_vector_type(8))) float v8f;
typedef __attribute__((ext_vector_type(4))) int   v4i;

typedef __attribute__((address_space(1))) v4i* g1_v4i;   // global (device) AS
typedef __attribute__((address_space(3))) v4i* l3_v4i;   // LDS AS

__device__ __forceinline__ void wait_async_then_barrier() {
#if __has_builtin(__builtin_amdgcn_s_wait_asynccnt)
    __builtin_amdgcn_s_wait_asynccnt(0);
#else
    asm volatile("s_wait_asynccnt 0" ::: "memory");
#endif
    __syncthreads();
}

// ---------------------------------------------------------------------------
// Kernel 1: QKV projection via V_WMMA_F32_16X16X4_F32.
// Block = 256 threads = 8 wave32s, handles a 64-row node tile (4 x 16-row
// WMMA tiles). Wave w owns output columns [16w,16w+16) for Q, K and V; each
// B-fragment (weight slice) is reused across the 4 row tiles and the A-
// fragment is reused across the 3 weight matrices -> 12 WMMAs per k-step.
// Fragment layouts per ISA 7.12.2 (32-bit):
//   A 16x4 : lane%16 = M, lane/16 selects K pair {0,1} vs {2,3} in (v0,v1)
//   B 4x16 : lane%16 = N, lane/16 selects K pair
//   C/D    : VGPR r -> M = r + 8*(lane/16), N = lane%16
// ---------------------------------------------------------------------------
__global__ __launch_bounds__(256) void qkv_wmma_kernel(
    const float* __restrict__ h,
    const float* __restrict__ WQ, const float* __restrict__ bQ,
    const float* __restrict__ WK, const float* __restrict__ bK,
    const float* __restrict__ WV, const float* __restrict__ bV,
    float* __restrict__ Q, float* __restrict__ K, float* __restrict__ V,
    int n_nodes)
{
    // padded stride 132 floats: conflict-free column reads, rows 16B aligned
    __shared__ float hTile[ROWS_PER_BLOCK][IN_DIM + 4];

    const int tile_row = blockIdx.x * ROWS_PER_BLOCK;
    const int tid      = threadIdx.x;

    // cooperative load: 64x128 floats = 2048 float4, 256 threads -> 8 each
    for (int i = tid; i < ROWS_PER_BLOCK * (IN_DIM / 4); i += 256) {
        int r  = i >> 5;          // float4-row
        int c4 = i & 31;
        int gr = tile_row + r;
        gr = gr < n_nodes ? gr : n_nodes - 1;   // branchless clamp
        const float4* gsrc = (const float4*)h + (size_t)gr * (IN_DIM / 4) + c4;
        float*        ldst = &hTile[r][c4 * 4];
#if __has_builtin(__builtin_amdgcn_global_load_async_to_lds_b128)
        // CDNA5 async DMA: global -> LDS directly, tracked by ASYNCcnt.
        // Builtin signature: (v4i AS1*, v4i AS3*, imm offset, imm cpol)
        __builtin_amdgcn_global_load_async_to_lds_b128(
            (g1_v4i)gsrc, (l3_v4i)ldst, 0, 0);
#else
        *(float4*)ldst = *gsrc;
#endif
    }
    wait_async_then_barrier();

    const int wave  = tid >> 5;        // 0..7 -> column tile
    const int lane  = tid & 31;
    const int l16   = lane & 15;       // M (A) / N (B, C, D)
    const int khalf = lane >> 4;       // K pair select
    const int col   = wave * 16 + l16; // global output column

    v8f accQ[4] = {}, accK[4] = {}, accV[4] = {};

    #pragma unroll 2
    for (int k0 = 0; k0 < IN_DIM; k0 += 4) {
        const int ka = k0 + 2 * khalf;

        v2f bq, bk, bv;
        bq.x = WQ[(size_t)(ka    ) * OUT_DIM + col];
        bq.y = WQ[(size_t)(ka + 1) * OUT_DIM + col];
        bk.x = WK[(size_t)(ka    ) * OUT_DIM + col];
        bk.y = WK[(size_t)(ka + 1) * OUT_DIM + col];
        bv.x = WV[(size_t)(ka    ) * OUT_DIM + col];
        bv.y = WV[(size_t)(ka + 1) * OUT_DIM + col];

        #pragma unroll
        for (int rt = 0; rt < 4; ++rt) {
            v2f a;
            a.x = hTile[rt * 16 + l16][ka];
            a.y = hTile[rt * 16 + l16][ka + 1];
            accQ[rt] = __builtin_amdgcn_wmma_f32_16x16x4_f32(false, a, false, bq, (short)0, accQ[rt], false, false);
            accK[rt] = __builtin_amdgcn_wmma_f32_16x16x4_f32(false, a, false, bk, (short)0, accK[rt], false, false);
            accV[rt] = __builtin_amdgcn_wmma_f32_16x16x4_f32(false, a, false, bv, (short)0, accV[rt], false, false);
        }
    }

    const float biasQ = bQ[col];
    const float biasK = bK[col];
    const float biasV = bV[col];

    #pragma unroll
    for (int rt = 0; rt < 4; ++rt) {
        #pragma unroll
        for (int r = 0; r < 8; ++r) {
            int row = tile_row + rt * 16 + r + 8 * khalf;
            if (row < n_nodes) {
                size_t o = (size_t)row * OUT_DIM + col;
                Q[o] = accQ[rt][r] + biasQ;
                K[o] = accK[rt][r] + biasK;
                V[o] = accV[rt][r] + biasV;
            }
        }
    }
}

// ---------------------------------------------------------------------------
// Kernel 2: per-(edge, head) attention score + scatter. 8 consecutive threads
// share one edge (one head each) so their float4 gathers cover one contiguous
// 512B row. Q/K/V (77MB) are L2-resident on MI455X (192MB L2); atomics land
// in the L2 atomic units.
// ---------------------------------------------------------------------------
__global__ __launch_bounds__(256) void edge_attn_kernel(
    const float* __restrict__ Q, const float* __restrict__ K,
    const float* __restrict__ V,
    const int* __restrict__ src, const int* __restrict__ dst,
    float* __restrict__ out, float* __restrict__ z, int n_edges)
{
    long long idx = (long long)blockIdx.x * 256 + threadIdx.x;
    int e  = (int)(idx >> 3);
    int hd = (int)(idx & 7);
    if (e >= n_edges) return;

    const int s = src[e];
    const int d = dst[e];

    const float4* K4 = (const float4*)(K + (size_t)s * OUT_DIM + hd * HEAD_DIM);
    const float4* Q4 = (const float4*)(Q + (size_t)d * OUT_DIM + hd * HEAD_DIM);
    const float4* V4 = (const float4*)(V + (size_t)s * OUT_DIM + hd * HEAD_DIM);

    float dot = 0.0f;
    #pragma unroll
    for (int j = 0; j < 4; ++j) {
        float4 kv = K4[j];
        float4 qv = Q4[j];
        dot += kv.x * qv.x + kv.y * qv.y + kv.z * qv.z + kv.w * qv.w;
    }

    // score / sqrt(HEAD_DIM) = score * 0.25, clamp [-5, 5], exp
    float w = __expf(fminf(fmaxf(dot * 0.25f, -5.0f), 5.0f));

    atomicAdd(&z[(size_t)d * HEADS + hd], w);

    float* o = out + (size_t)d * OUT_DIM + hd * HEAD_DIM;
    #pragma unroll
    for (int j = 0; j < 4; ++j) {
        float4 vv = V4[j];
        atomicAdd(&o[j * 4 + 0], w * vv.x);
        atomicAdd(&o[j * 4 + 1], w * vv.y);
        atomicAdd(&o[j * 4 + 2], w * vv.z);
        atomicAdd(&o[j * 4 + 3], w * vv.w);
    }
}

// ---------------------------------------------------------------------------
// Kernel 3: normalize by segment-softmax denominator where z > 0.
// out accumulated wV; if z == 0 there were no edges -> out stays 0 (memset).
// ---------------------------------------------------------------------------
__global__ __launch_bounds__(256) void finalize_kernel(
    float* __restrict__ out, const float* __restrict__ z, int n_nodes)
{
    long long idx = (long long)blockIdx.x * 256 + threadIdx.x;
    if (idx >= (long long)n_nodes * OUT_DIM) return;
    int n  = (int)(idx >> 7);           // / 128
    int hd = (int)((idx >> 4) & 7);     // head
    float zv = z[(size_t)n * HEADS + hd];
    if (zv > 0.0f) out[idx] /= zv;
}

// ---------------------------------------------------------------------------
extern "C" void kernel_launch(void* const* d_in, const int* in_sizes, int n_in,
                              void* d_out, int out_size, void* d_ws, size_t ws_size,
                              hipStream_t stream) {
    const float* h   = (const float*)d_in[0];
    const int*   src = (const int*)  d_in[1];
    const int*   dst = (const int*)  d_in[2];
    const float* WQ  = (const float*)d_in[3];
    const float* bQ  = (const float*)d_in[4];
    const float* WK  = (const float*)d_in[5];
    const float* bK  = (const float*)d_in[6];
    const float* WV  = (const float*)d_in[7];
    const float* bV  = (const float*)d_in[8];

    const int n_nodes = in_sizes[0] / IN_DIM;
    const int n_edges = in_sizes[1];

    // workspace: Q | K | V | z   (f32)
    float* Q = (float*)d_ws;
    float* K = Q + (size_t)n_nodes * OUT_DIM;
    float* V = K + (size_t)n_nodes * OUT_DIM;
    float* z = V + (size_t)n_nodes * OUT_DIM;
    float* out = (float*)d_out;

    (void)hipMemsetAsync(out, 0, (size_t)n_nodes * OUT_DIM * sizeof(float), stream);
    (void)hipMemsetAsync(z,   0, (size_t)n_nodes * HEADS   * sizeof(float), stream);

    const int row_tiles = (n_nodes + ROWS_PER_BLOCK - 1) / ROWS_PER_BLOCK;
    qkv_wmma_kernel<<<row_tiles, 256, 0, stream>>>(h, WQ, bQ, WK, bK, WV, bV,
                                                   Q, K, V, n_nodes);

    const long long eh = (long long)n_edges * HEADS;
    edge_attn_kernel<<<(unsigned)((eh + 255) / 256), 256, 0, stream>>>(
        Q, K, V, src, dst, out, z, n_edges);

    const long long no = (long long)n_nodes * OUT_DIM;
    finalize_kernel<<<(unsigned)((no + 255) / 256), 256, 0, stream>>>(out, z, n_nodes);
}